// ClusterGNN_35923106463765
// MI455X (gfx1250) — compile-verified
//
#include <hip/hip_runtime.h>

// ---------------------------------------------------------------------------
// ClusterGNN forward for MI455X (gfx1250, wave32).
//
// Exact simplification: s = softmax(softmax(pool,axis=0), axis=-1) on [N,1]
// is identically 1.0, so emb = column-sum of g2; Wlp/blp/Wrp are dead.
//
// GEMMs: V_WMMA_F32_16X16X4_F32, one wave per 32x64 output tile (2x4
// register blocking -> 8 WMMAs per 6 b64 loads).  Weights are pre-packed
// once per launch into (W[k][n],W[k+1][n]) pairs so every B fragment is one
// aligned global_load_b64.  Partial M-tiles handled by CLAMPING A-row
// addresses (C row m depends only on A row m; clamped duplicate rows are
// never stored) -- K-loops are fully branchless.
//
// Scatter-mean: global_atomic_add_f32 into L2-resident agg (~5.2 GB atomic
// traffic total; 192 MB L2 holds the whole working set).
// ---------------------------------------------------------------------------

#define B_   16
#define N_   5000
#define E_   160000
#define DIN  128
#define DENC 256
#define DGNN 256
#define DFC  128
#define NCLS 2

#define MT32 ((N_ + 31) / 32)      // 157 M-tiles of 32 rows

typedef float v2f __attribute__((ext_vector_type(2)));
typedef float v8f __attribute__((ext_vector_type(8)));

__device__ __forceinline__ int clamp_row(int r) { return r < N_ ? r : (N_ - 1); }
__device__ __forceinline__ float relu_(float v) { return v > 0.0f ? v : 0.0f; }

#define WMMA_F32(A, Bm, C) \
    __builtin_amdgcn_wmma_f32_16x16x4_f32(false, (A), false, (Bm), (short)0, (C), false, false)

// ---------------------------------------------------------------------------
// Pack W[K,Ncols] -> Wp[K/2, Ncols, 2] with Wp[p][n] = (W[2p][n], W[2p+1][n]).
// Makes the WMMA B fragment (rows kk, kk+1 of column n) one b64 load.
// ---------------------------------------------------------------------------
__global__ void k_pack(const float* __restrict__ W, float* __restrict__ Wp,
                       int Khalf, int Ncols)
{
    const int i = blockIdx.x * blockDim.x + threadIdx.x;
    if (i < Khalf * Ncols) {
        const int p = i / Ncols, n = i % Ncols;
        Wp[2 * i + 0] = W[(size_t)(2 * p)     * Ncols + n];
        Wp[2 * i + 1] = W[(size_t)(2 * p + 1) * Ncols + n];
    }
}

// ---------------------------------------------------------------------------
// Encoder: h = relu(x @ We + be)   x:[N,128]  Wep: packed We [64,256,2]
// A (16x4 f32) per ISA 7.12.2: lane 0-15 = M, half-wave = K offset +2,
// VGPR = K +1.  B (4x16) symmetric.  C/D: VGPR r = C[r + 8*hi][lane&15].
// ---------------------------------------------------------------------------
__global__ __launch_bounds__(32)
void k_encoder(const float* __restrict__ x, const float* __restrict__ Wep,
               const float* __restrict__ be, float* __restrict__ h)
{
    const int lane = threadIdx.x;
    int t = blockIdx.x;
    const int nt = t % (DENC / 64); t /= (DENC / 64);
    const int mt = t % MT32;        t /= MT32;
    const int b  = t;

    const int m0 = mt * 32, n0 = nt * 64;
    const int lm = lane & 15;
    const int hi = lane >> 4;            // 0 or 1

    const float* xb  = x + (size_t)b * N_ * DIN;
    const float* pa0 = xb + (size_t)clamp_row(m0 + lm)      * DIN;
    const float* pa1 = xb + (size_t)clamp_row(m0 + 16 + lm) * DIN;

    v8f c00 = {}, c01 = {}, c02 = {}, c03 = {};
    v8f c10 = {}, c11 = {}, c12 = {}, c13 = {};
#pragma unroll 2
    for (int k = 0; k < DIN; k += 4) {
        const int kk = k + hi * 2;
        const v2f a0 = *(const v2f*)(pa0 + kk);                       // b64
        const v2f a1 = *(const v2f*)(pa1 + kk);
        const float* wp = Wep + ((size_t)(kk >> 1) * DENC + n0 + lm) * 2;
        const v2f b0 = *(const v2f*)(wp);                             // b64
        const v2f b1 = *(const v2f*)(wp + 32);                        // +16 cols
        const v2f b2 = *(const v2f*)(wp + 64);
        const v2f b3 = *(const v2f*)(wp + 96);
        c00 = WMMA_F32(a0, b0, c00);  c01 = WMMA_F32(a0, b1, c01);
        c02 = WMMA_F32(a0, b2, c02);  c03 = WMMA_F32(a0, b3, c03);
        c10 = WMMA_F32(a1, b0, c10);  c11 = WMMA_F32(a1, b1, c11);
        c12 = WMMA_F32(a1, b2, c12);  c13 = WMMA_F32(a1, b3, c13);
    }

    float* hb = h + (size_t)b * N_ * DENC;
    const int na = n0 + lm;
    const float bi0 = be[na], bi1 = be[na + 16], bi2 = be[na + 32], bi3 = be[na + 48];
#pragma unroll
    for (int r = 0; r < 8; ++r) {
        const int row0 = m0 + r + hi * 8;
        const int row1 = row0 + 16;
        if (row0 < N_) {
            float* p = hb + (size_t)row0 * DENC + na;
            p[0]  = relu_(c00[r] + bi0);
            p[16] = relu_(c01[r] + bi1);
            p[32] = relu_(c02[r] + bi2);
            p[48] = relu_(c03[r] + bi3);
        }
        if (row1 < N_) {
            float* p = hb + (size_t)row1 * DENC + na;
            p[0]  = relu_(c10[r] + bi0);
            p[16] = relu_(c11[r] + bi1);
            p[32] = relu_(c12[r] + bi2);
            p[48] = relu_(c13[r] + bi3);
        }
    }
}

// ---------------------------------------------------------------------------
// SAGE layer: out = relu( (agg/max(deg,1)) @ Wl + bl + xin @ Wr )
// Two branchless K=256 passes accumulating into the same 2x4 tile block.
// Wlp/Wrp are packed [128,256,2].
// ---------------------------------------------------------------------------
__global__ __launch_bounds__(32)
void k_sage(const float* __restrict__ agg, const float* __restrict__ deg,
            const float* __restrict__ xin, const float* __restrict__ Wlp,
            const float* __restrict__ bl,  const float* __restrict__ Wrp,
            float* __restrict__ out)
{
    const int lane = threadIdx.x;
    int t = blockIdx.x;
    const int nt = t % (DGNN / 64); t /= (DGNN / 64);
    const int mt = t % MT32;        t /= MT32;
    const int b  = t;

    const int m0 = mt * 32, n0 = nt * 64;
    const int lm = lane & 15;
    const int hi = lane >> 4;

    const int r0 = clamp_row(m0 + lm);
    const int r1 = clamp_row(m0 + 16 + lm);
    const float* aggb = agg + (size_t)b * N_ * DENC;
    const float* xinb = xin + (size_t)b * N_ * DENC;
    const float inv0 = 1.0f / fmaxf(deg[(size_t)b * N_ + r0], 1.0f);
    const float inv1 = 1.0f / fmaxf(deg[(size_t)b * N_ + r1], 1.0f);

    v8f c00 = {}, c01 = {}, c02 = {}, c03 = {};
    v8f c10 = {}, c11 = {}, c12 = {}, c13 = {};

    // pass 1: mean @ Wl
    {
        const float* pa0 = aggb + (size_t)r0 * DENC;
        const float* pa1 = aggb + (size_t)r1 * DENC;
#pragma unroll 2
        for (int k = 0; k < DENC; k += 4) {
            const int kk = k + hi * 2;
            v2f a0 = *(const v2f*)(pa0 + kk);
            v2f a1 = *(const v2f*)(pa1 + kk);
            a0.x *= inv0; a0.y *= inv0;
            a1.x *= inv1; a1.y *= inv1;
            const float* wp = Wlp + ((size_t)(kk >> 1) * DGNN + n0 + lm) * 2;
            const v2f b0 = *(const v2f*)(wp);
            const v2f b1 = *(const v2f*)(wp + 32);
            const v2f b2 = *(const v2f*)(wp + 64);
            const v2f b3 = *(const v2f*)(wp + 96);
            c00 = WMMA_F32(a0, b0, c00);  c01 = WMMA_F32(a0, b1, c01);
            c02 = WMMA_F32(a0, b2, c02);  c03 = WMMA_F32(a0, b3, c03);
            c10 = WMMA_F32(a1, b0, c10);  c11 = WMMA_F32(a1, b1, c11);
            c12 = WMMA_F32(a1, b2, c12);  c13 = WMMA_F32(a1, b3, c13);
        }
    }
    // pass 2: xin @ Wr (accumulate into same C)
    {
        const float* pa0 = xinb + (size_t)r0 * DENC;
        const float* pa1 = xinb + (size_t)r1 * DENC;
#pragma unroll 2
        for (int k = 0; k < DENC; k += 4) {
            const int kk = k + hi * 2;
            const v2f a0 = *(const v2f*)(pa0 + kk);
            const v2f a1 = *(const v2f*)(pa1 + kk);
            const float* wp = Wrp + ((size_t)(kk >> 1) * DGNN + n0 + lm) * 2;
            const v2f b0 = *(const v2f*)(wp);
            const v2f b1 = *(const v2f*)(wp + 32);
            const v2f b2 = *(const v2f*)(wp + 64);
            const v2f b3 = *(const v2f*)(wp + 96);
            c00 = WMMA_F32(a0, b0, c00);  c01 = WMMA_F32(a0, b1, c01);
            c02 = WMMA_F32(a0, b2, c02);  c03 = WMMA_F32(a0, b3, c03);
            c10 = WMMA_F32(a1, b0, c10);  c11 = WMMA_F32(a1, b1, c11);
            c12 = WMMA_F32(a1, b2, c12);  c13 = WMMA_F32(a1, b3, c13);
        }
    }

    float* ob = out + (size_t)b * N_ * DGNN;
    const int na = n0 + lm;
    const float bi0 = bl[na], bi1 = bl[na + 16], bi2 = bl[na + 32], bi3 = bl[na + 48];
#pragma unroll
    for (int r = 0; r < 8; ++r) {
        const int row0 = m0 + r + hi * 8;
        const int row1 = row0 + 16;
        if (row0 < N_) {
            float* p = ob + (size_t)row0 * DGNN + na;
            p[0]  = relu_(c00[r] + bi0);
            p[16] = relu_(c01[r] + bi1);
            p[32] = relu_(c02[r] + bi2);
            p[48] = relu_(c03[r] + bi3);
        }
        if (row1 < N_) {
            float* p = ob + (size_t)row1 * DGNN + na;
            p[0]  = relu_(c10[r] + bi0);
            p[16] = relu_(c11[r] + bi1);
            p[32] = relu_(c12[r] + bi2);
            p[48] = relu_(c13[r] + bi3);
        }
    }
}

// ---------------------------------------------------------------------------
// Zero-fill scratch (agg / deg).
// ---------------------------------------------------------------------------
__global__ void k_zero(float* __restrict__ p, size_t n)
{
    size_t i = (size_t)blockIdx.x * blockDim.x + threadIdx.x;
    const size_t stride = (size_t)gridDim.x * blockDim.x;
    for (; i < n; i += stride) p[i] = 0.0f;
}

// ---------------------------------------------------------------------------
// Edge scatter: agg[dst] += feat[src] (per feature), deg[dst] += 1 (once).
// One 256-thread block per (bag, edge); thread = feature.
// ---------------------------------------------------------------------------
__global__ __launch_bounds__(256)
void k_scatter(const float* __restrict__ feat, const long long* __restrict__ ei,
               float* __restrict__ agg, float* __restrict__ deg, int count_deg)
{
    const int e = blockIdx.x % E_;
    const int b = blockIdx.x / E_;
    const long long* eib = ei + (size_t)b * 2 * E_;
    const int src = (int)eib[e];
    const int dst = (int)eib[E_ + e];
    const int f = threadIdx.x;

    const float v = feat[((size_t)b * N_ + src) * DENC + f];
    atomicAdd(&agg[((size_t)b * N_ + dst) * DENC + f], v);
    if (count_deg && f == 0)
        atomicAdd(&deg[(size_t)b * N_ + dst], 1.0f);
}

// ---------------------------------------------------------------------------
// emb[b,f] = sum_n g[b,n,f]   (diff-pool with 1 cluster; s == ones exactly)
// ---------------------------------------------------------------------------
__global__ __launch_bounds__(256)
void k_colsum(const float* __restrict__ g, float* __restrict__ emb)
{
    const int b = blockIdx.x;
    const int f = threadIdx.x;
    const float* gb = g + (size_t)b * N_ * DGNN;
    float s = 0.0f;
    for (int n = 0; n < N_; ++n) {
        __builtin_prefetch(gb + (size_t)(n + 16) * DGNN + f, 0, 0);
        s += gb[(size_t)n * DGNN + f];
    }
    emb[b * DGNN + f] = s;
}

// ---------------------------------------------------------------------------
// Classifier: out = relu(emb @ Wc1 + bc1) @ Wc2 + bc2.  One block per bag.
// ---------------------------------------------------------------------------
__global__ __launch_bounds__(128)
void k_cls(const float* __restrict__ emb, const float* __restrict__ Wc1,
           const float* __restrict__ bc1, const float* __restrict__ Wc2,
           const float* __restrict__ bc2, float* __restrict__ out)
{
    __shared__ float h1[DFC];
    __shared__ float red[DFC];
    const int b = blockIdx.x;
    const int j = threadIdx.x;
    const float* e = emb + b * DGNN;

    float acc = bc1[j];
    for (int k = 0; k < DGNN; ++k)
        acc += e[k] * Wc1[(size_t)k * DFC + j];
    h1[j] = acc > 0.0f ? acc : 0.0f;
    __syncthreads();

    for (int cidx = 0; cidx < NCLS; ++cidx) {
        red[j] = h1[j] * Wc2[(size_t)j * NCLS + cidx];
        __syncthreads();
        for (int s = 64; s > 0; s >>= 1) {
            if (j < s) red[j] += red[j + s];
            __syncthreads();
        }
        if (j == 0) out[b * NCLS + cidx] = red[0] + bc2[cidx];
        __syncthreads();
    }
}

// ---------------------------------------------------------------------------
extern "C" void kernel_launch(void* const* d_in, const int* in_sizes, int n_in,
                              void* d_out, int out_size, void* d_ws, size_t ws_size,
                              hipStream_t stream)
{
    const float*     x   = (const float*)    d_in[0];
    const long long* ei  = (const long long*)d_in[1];   // int64 per reference
    const float*     We  = (const float*)    d_in[2];
    const float*     be  = (const float*)    d_in[3];
    const float*     Wl1 = (const float*)    d_in[4];
    const float*     bl1 = (const float*)    d_in[5];
    const float*     Wr1 = (const float*)    d_in[6];
    const float*     Wl2 = (const float*)    d_in[7];
    const float*     bl2 = (const float*)    d_in[8];
    const float*     Wr2 = (const float*)    d_in[9];
    // d_in[10..12] (Wlp, blp, Wrp) provably do not affect the output.
    const float*     Wc1 = (const float*)    d_in[13];
    const float*     bc1 = (const float*)    d_in[14];
    const float*     Wc2 = (const float*)    d_in[15];
    const float*     bc2 = (const float*)    d_in[16];

    // workspace (floats): h | g1 | agg | deg | emb | packed weights ; g2 = h.
    const size_t feat = (size_t)B_ * N_ * DENC;
    float* h    = (float*)d_ws;
    float* g1   = h    + feat;
    float* agg  = g1   + feat;
    float* deg  = agg  + feat;
    float* emb  = deg  + (size_t)B_ * N_;
    float* Wep  = emb  + (size_t)B_ * DGNN;
    float* Wl1p = Wep  + (size_t)DIN  * DENC;
    float* Wr1p = Wl1p + (size_t)DENC * DGNN;
    float* Wl2p = Wr1p + (size_t)DENC * DGNN;
    float* Wr2p = Wl2p + (size_t)DENC * DGNN;
    float* g2   = h;                      // h is dead after SAGE layer 1

    const dim3 wave(32);
    const int  gemmGrid = B_ * MT32 * (DENC / 64);   // 16*157*4 = 10048 waves

    // pack weights: (W[k][n], W[k+1][n]) contiguous -> B fragment = 1 b64 load
    k_pack<<<(DIN / 2 * DENC + 255) / 256, 256, 0, stream>>>(We,  Wep,  DIN / 2,  DENC);
    k_pack<<<(DENC / 2 * DGNN + 255) / 256, 256, 0, stream>>>(Wl1, Wl1p, DENC / 2, DGNN);
    k_pack<<<(DENC / 2 * DGNN + 255) / 256, 256, 0, stream>>>(Wr1, Wr1p, DENC / 2, DGNN);
    k_pack<<<(DENC / 2 * DGNN + 255) / 256, 256, 0, stream>>>(Wl2, Wl2p, DENC / 2, DGNN);
    k_pack<<<(DENC / 2 * DGNN + 255) / 256, 256, 0, stream>>>(Wr2, Wr2p, DENC / 2, DGNN);

    // encoder
    k_encoder<<<gemmGrid, wave, 0, stream>>>(x, Wep, be, h);

    // SAGE layer 1
    k_zero<<<2048, 256, 0, stream>>>(agg, feat);
    k_zero<<<256, 256, 0, stream>>>(deg, (size_t)B_ * N_);
    k_scatter<<<B_ * E_, 256, 0, stream>>>(h, ei, agg, deg, 1);
    k_sage<<<gemmGrid, wave, 0, stream>>>(agg, deg, h, Wl1p, bl1, Wr1p, g1);

    // SAGE layer 2 (deg identical; only re-zero agg)
    k_zero<<<2048, 256, 0, stream>>>(agg, feat);
    k_scatter<<<B_ * E_, 256, 0, stream>>>(g1, ei, agg, deg, 0);
    k_sage<<<gemmGrid, wave, 0, stream>>>(agg, deg, g1, Wl2p, bl2, Wr2p, g2);

    // pooling (s == ones) + classifier
    k_colsum<<<B_, 256, 0, stream>>>(g2, emb);
    k_cls<<<B_, 128, 0, stream>>>(emb, Wc1, bc1, Wc2, bc2, (float*)d_out);
}